// HeteroFraudGNN_88373247082632
// MI455X (gfx1250) — compile-verified
//
#include <hip/hip_runtime.h>
#include <hip/hip_bf16.h>

typedef __attribute__((ext_vector_type(2))) float v2f;
typedef __attribute__((ext_vector_type(8))) float v8f;

// ---------------------------------------------------------------------------
// Fused WMMA GEMM:  C[M x N] = (A1 * inv_cnt?) @ W1 + (A2 @ W2)? + bias  (+= C?)
// K = KS*4, N = NT*16.  f32 WMMA 16x16x4, weights staged in LDS.
// One wave -> one 16-row M tile, NT accumulators of 16x16.
// ---------------------------------------------------------------------------
template <int KS, int NT, bool HAS_A2, bool HAS_CNT, bool ACC>
__global__ __launch_bounds__(256) void gemm_wmma(
    const float* __restrict__ A1, const float* __restrict__ cnt,
    const float* __restrict__ W1g, const float* __restrict__ A2,
    const float* __restrict__ W2g, const float* __restrict__ bias,
    float* __restrict__ C, int M) {
  constexpr int K = KS * 4;
  constexpr int N = NT * 16;
  __shared__ float sW1[K * N];
  __shared__ float sW2[HAS_A2 ? K * N : 1];

  const int tid = threadIdx.x;
  for (int i = tid; i < K * N; i += blockDim.x) sW1[i] = W1g[i];
  if (HAS_A2)
    for (int i = tid; i < K * N; i += blockDim.x) sW2[i] = W2g[i];
  __syncthreads();

  const int wave = tid >> 5;
  const int lane = tid & 31;
  const int m0 = (blockIdx.x * (int)(blockDim.x >> 5) + wave) * 16;
  if (m0 >= M) return;  // wave-uniform: EXEC stays all-ones for WMMA

  const int row = m0 + (lane & 15);
  const int khalf = (lane >> 4) * 2;  // 0 for lanes 0-15, 2 for lanes 16-31
  const int ncol = lane & 15;

  float inv = 1.0f;
  if (HAS_CNT) inv = 1.0f / fmaxf(cnt[row], 1.0f);

  v8f acc[NT] = {};
  const float* a1p = A1 + (size_t)row * K + khalf;
  const float* a2p = HAS_A2 ? (A2 + (size_t)row * K + khalf) : nullptr;

  for (int s = 0; s < KS; ++s) {
    const int k0 = s * 4;
    v2f a1;
    a1.x = a1p[k0];
    a1.y = a1p[k0 + 1];
    if (HAS_CNT) { a1.x *= inv; a1.y *= inv; }
    v2f a2;
    if (HAS_A2) { a2.x = a2p[k0]; a2.y = a2p[k0 + 1]; }
#pragma unroll
    for (int t = 0; t < NT; ++t) {
      v2f b1;
      b1.x = sW1[(k0 + khalf) * N + t * 16 + ncol];
      b1.y = sW1[(k0 + khalf + 1) * N + t * 16 + ncol];
      acc[t] = __builtin_amdgcn_wmma_f32_16x16x4_f32(false, a1, false, b1,
                                                     (short)0, acc[t], false, false);
      if (HAS_A2) {
        v2f b2;
        b2.x = sW2[(k0 + khalf) * N + t * 16 + ncol];
        b2.y = sW2[(k0 + khalf + 1) * N + t * 16 + ncol];
        acc[t] = __builtin_amdgcn_wmma_f32_16x16x4_f32(false, a2, false, b2,
                                                       (short)0, acc[t], false, false);
      }
    }
  }

  // C/D layout: VGPR r -> lanes 0-15: (M=m0+r, N=t*16+lane); lanes 16-31: M=m0+8+r
  const int mbase = m0 + (lane >> 4) * 8;
#pragma unroll
  for (int t = 0; t < NT; ++t) {
    const float bv = bias[t * 16 + ncol];
#pragma unroll
    for (int r = 0; r < 8; ++r) {
      const size_t idx = (size_t)(mbase + r) * N + t * 16 + ncol;
      float v = acc[t][r] + bv;
      if (ACC) v += C[idx];
      C[idx] = v;
    }
  }
}

// ---------------------------------------------------------------------------
// Scatter-mean accumulation: agg[dst] += h[src] (64 f32 via float4 atomics),
// cnt[dst] += 1.  16 threads/edge, 4 cols each.
// ---------------------------------------------------------------------------
__global__ void scatter_kernel(const int* __restrict__ edge, long E,
                               const float* __restrict__ hsrc,
                               float* __restrict__ agg, float* __restrict__ cnt) {
  long t = (long)blockIdx.x * blockDim.x + threadIdx.x;
  const long total = E * 16;
  const long stride = (long)gridDim.x * blockDim.x;
  for (; t < total; t += stride) {
    const long e = t >> 4;
    const int part = (int)(t & 15);
    const int s = edge[e];
    const int d = edge[E + e];
    const float4 v = *(const float4*)(hsrc + (size_t)s * 64 + part * 4);
    float* o = agg + (size_t)d * 64 + part * 4;
    atomicAdd(o + 0, v.x);
    atomicAdd(o + 1, v.y);
    atomicAdd(o + 2, v.z);
    atomicAdd(o + 3, v.w);
    if (part == 0) atomicAdd(&cnt[d], 1.0f);
  }
}

// ---------------------------------------------------------------------------
// BN stats: stats[0:C]=sum, stats[C:2C]=sumsq over rows (block LDS reduce ->
// global atomics). blockDim must be a multiple of C and >= 2C.
// ---------------------------------------------------------------------------
template <int C>
__global__ void bn_stats_kernel(const float* __restrict__ x, int n,
                                float* __restrict__ stats) {
  __shared__ float ssum[C], ssq[C];
  const int tid = threadIdx.x;
  if (tid < C) { ssum[tid] = 0.f; ssq[tid] = 0.f; }
  __syncthreads();
  const int col = tid % C;
  float s = 0.f, q = 0.f;
  const long total = (long)n * C;
  const long stride = (long)gridDim.x * blockDim.x;
  for (long i = (long)blockIdx.x * blockDim.x + tid; i < total; i += stride) {
    const float v = x[i];
    s += v;
    q += v * v;
  }
  atomicAdd(&ssum[col], s);
  atomicAdd(&ssq[col], q);
  __syncthreads();
  if (tid < C) atomicAdd(&stats[tid], ssum[tid]);
  else if (tid < 2 * C) atomicAdd(&stats[tid], ssq[tid - C]);
}

template <int C>
__global__ void bn_apply_kernel(const float* __restrict__ src, float* __restrict__ dst,
                                int n, const float* __restrict__ stats,
                                const float* __restrict__ g, const float* __restrict__ b) {
  const long total = (long)n * C;
  const long stride = (long)gridDim.x * blockDim.x;
  const float invn = 1.0f / (float)n;
  for (long i = (long)blockIdx.x * blockDim.x + threadIdx.x; i < total; i += stride) {
    const int col = (int)(i % C);
    const float m = stats[col] * invn;
    const float var = stats[C + col] * invn - m * m;
    float y = (src[i] - m) * rsqrtf(var + 1e-5f) * g[col] + b[col];
    dst[i] = fmaxf(y, 0.0f);
  }
}

__global__ void zero_kernel(float* __restrict__ p, long n) {
  const long stride = (long)gridDim.x * blockDim.x;
  for (long i = (long)blockIdx.x * blockDim.x + threadIdx.x; i < n; i += stride)
    p[i] = 0.0f;
}

// out[i] = sigmoid(h2[i,:] @ W3 + b3),  h2: [n,32]
__global__ void head_kernel(const float* __restrict__ h2, const float* __restrict__ W3,
                            const float* __restrict__ b3, float* __restrict__ out, int n) {
  const int i = blockIdx.x * blockDim.x + threadIdx.x;
  if (i >= n) return;
  float s = b3[0];
  const float* r = h2 + (size_t)i * 32;
#pragma unroll
  for (int k = 0; k < 32; ++k) s += r[k] * W3[k];
  out[i] = 1.0f / (1.0f + expf(-s));
}

// ---------------------------------------------------------------------------
extern "C" void kernel_launch(void* const* d_in, const int* in_sizes, int n_in,
                              void* d_out, int out_size, void* d_ws, size_t ws_size,
                              hipStream_t stream) {
  // node types: 0 customer, 1 transaction, 2 device, 3 email, 4 address
  static const int Nn[5] = {200000, 1000000, 100000, 150000, 120000};
  static const int INn[5] = {16, 32, 8, 8, 8};
  // position in alphabetically-sorted key order (jax pytree flatten)
  static const int posNT[5] = {1, 4, 2, 3, 0};
  // edge types in input order: makes, used_in, linked_to, located_at,
  //                            rev_makes, rev_used_in, rev_linked_to, rev_located_at
  static const int posET[8] = {2, 7, 0, 1, 5, 6, 3, 4};  // alpha-sorted position
  static const int esrc[8] = {0, 2, 3, 4, 1, 1, 1, 1};
  static const int edst[8] = {1, 1, 1, 1, 0, 2, 3, 4};

  const float* x[5];
  for (int i = 0; i < 5; ++i) x[i] = (const float*)d_in[i];
  const int* edge[8];
  long Ee[8];
  for (int i = 0; i < 8; ++i) { edge[i] = (const int*)d_in[5 + i]; Ee[i] = in_sizes[5 + i] / 2; }

  auto P = [&](int i) { return (const float*)d_in[i]; };
  // param leaf indices (base 13), dict keys sorted:
  // bn1: 13..22 (address{b,g}, customer, device, email, transaction)
  // bn2: 23..32
  // cls: 33 W1, 34 W2, 35 W3, 36 b1, 37 b2, 38 b3, 39 be1, 40 be2, 41 g1, 42 g2
  // conv1: 43..66  (per sorted edge: Wl, Wr, bl)
  // conv2: 67..90
  // proj: 91..110  (per sorted node: W, b, be, g)

  // workspace layout (floats)
  float* ws = (float*)d_ws;
  size_t off = 0;
  float* h[5];
  for (int i = 0; i < 5; ++i) { h[i] = ws + off; off += (size_t)Nn[i] * 64; }
  float* ob[5];
  for (int i = 0; i < 5; ++i) { ob[i] = ws + off; off += (size_t)Nn[i] * 64; }
  float* agg = ws + off; off += (size_t)1000000 * 64;
  float* cnt = ws + off; off += 1000000;
  float* stats = ws + off; off += 128;

  auto zero = [&](float* p, size_t n) {
    long blocks = (long)((n + 255) / 256);
    if (blocks > 16384) blocks = 16384;
    zero_kernel<<<dim3((unsigned)blocks), dim3(256), 0, stream>>>(p, (long)n);
  };

  // ---- input projection + BN + ReLU per node type ----
  for (int nt = 0; nt < 5; ++nt) {
    const int pw = 91 + posNT[nt] * 4;
    const int M = Nn[nt];
    const unsigned grid = (unsigned)((M + 127) / 128);
    if (INn[nt] == 8)
      gemm_wmma<2, 4, false, false, false><<<grid, 256, 0, stream>>>(
          x[nt], nullptr, P(pw), nullptr, nullptr, P(pw + 1), h[nt], M);
    else if (INn[nt] == 16)
      gemm_wmma<4, 4, false, false, false><<<grid, 256, 0, stream>>>(
          x[nt], nullptr, P(pw), nullptr, nullptr, P(pw + 1), h[nt], M);
    else
      gemm_wmma<8, 4, false, false, false><<<grid, 256, 0, stream>>>(
          x[nt], nullptr, P(pw), nullptr, nullptr, P(pw + 1), h[nt], M);
    zero(stats, 128);
    bn_stats_kernel<64><<<1024, 256, 0, stream>>>(h[nt], M, stats);
    bn_apply_kernel<64><<<2048, 256, 0, stream>>>(h[nt], h[nt], M, stats,
                                                  P(pw + 3), P(pw + 2));
  }

  // ---- two hetero SAGE layers ----
  for (int L = 0; L < 2; ++L) {
    const int cw = (L == 0) ? 43 : 67;
    const int bnb = (L == 0) ? 13 : 23;
    for (int nt = 0; nt < 5; ++nt) zero(ob[nt], (size_t)Nn[nt] * 64);
    for (int et = 0; et < 8; ++et) {
      const int s = esrc[et], d = edst[et];
      const int Md = Nn[d];
      zero(agg, (size_t)Md * 64);
      zero(cnt, (size_t)Md);
      scatter_kernel<<<4096, 256, 0, stream>>>(edge[et], Ee[et], h[s], agg, cnt);
      const int pe = cw + posET[et] * 3;
      gemm_wmma<16, 4, true, true, true><<<(unsigned)((Md + 127) / 128), 256, 0, stream>>>(
          agg, cnt, P(pe) /*Wl*/, h[d], P(pe + 1) /*Wr*/, P(pe + 2) /*bl*/, ob[d], Md);
    }
    for (int nt = 0; nt < 5; ++nt) {
      zero(stats, 128);
      bn_stats_kernel<64><<<1024, 256, 0, stream>>>(ob[nt], Nn[nt], stats);
      bn_apply_kernel<64><<<2048, 256, 0, stream>>>(ob[nt], h[nt], Nn[nt], stats,
                                                    P(bnb + posNT[nt] * 2 + 1),
                                                    P(bnb + posNT[nt] * 2));
    }
  }

  // ---- classifier head on transaction ----
  const int M = Nn[1];
  float* h1 = agg;     // 1M x 64 (agg region free now)
  float* h2 = ob[1];   // 1M x 32 (out-transaction region free now)
  const unsigned gM = (unsigned)((M + 127) / 128);
  gemm_wmma<16, 4, false, false, false><<<gM, 256, 0, stream>>>(
      h[1], nullptr, P(33), nullptr, nullptr, P(36), h1, M);
  zero(stats, 128);
  bn_stats_kernel<64><<<1024, 256, 0, stream>>>(h1, M, stats);
  bn_apply_kernel<64><<<2048, 256, 0, stream>>>(h1, h1, M, stats, P(41), P(39));

  gemm_wmma<16, 2, false, false, false><<<gM, 256, 0, stream>>>(
      h1, nullptr, P(34), nullptr, nullptr, P(37), h2, M);
  zero(stats, 128);
  bn_stats_kernel<32><<<1024, 256, 0, stream>>>(h2, M, stats);
  bn_apply_kernel<32><<<2048, 256, 0, stream>>>(h2, h2, M, stats, P(42), P(40));

  head_kernel<<<(unsigned)((M + 255) / 256), 256, 0, stream>>>(
      h2, P(35), P(38), (float*)d_out, M);
}